// DGFU_90838558310687
// MI455X (gfx1250) — compile-verified
//
#include <hip/hip_runtime.h>

#define Bn 8
#define Cn 256
#define Hn 128
#define Wn 128
#define HWn (Hn * Wn)
#define Kn 64

typedef __attribute__((ext_vector_type(2))) float v2f;
typedef __attribute__((ext_vector_type(8))) float v8f;
typedef __attribute__((ext_vector_type(4))) unsigned int v4u;
typedef __attribute__((ext_vector_type(8))) int v8i_;
typedef __attribute__((ext_vector_type(4))) int v4i_;

#define HAVE_TDM __has_builtin(__builtin_amdgcn_tensor_load_to_lds)

// ---------------------------------------------------------------- zero scratch
__global__ __launch_bounds__(256) void zero_kernel(float* p, int n) {
    int i = blockIdx.x * 256 + threadIdx.x;
    if (i < n) p[i] = 0.0f;
}

// ------------------------------------------------- cluster sums + counts (LDS)
#define PIX 1024
__global__ __launch_bounds__(256) void scatter_kernel(const float* __restrict__ x,
                                                      const int* __restrict__ index,
                                                      float* __restrict__ sums,
                                                      float* __restrict__ counts) {
    __shared__ float ls[128 * Kn];   // 32 KB: [c_chunk][k], bank = k
    __shared__ float lc[Kn];
    const int tid = threadIdx.x;
    const int b = blockIdx.x;
    const int hw0 = blockIdx.y * PIX;

    if (tid < Kn) lc[tid] = 0.0f;
    int idxreg[PIX / 256];
#pragma unroll
    for (int g = 0; g < PIX / 256; ++g)
        idxreg[g] = index[b * HWn + hw0 + g * 256 + tid];
    __syncthreads();
#pragma unroll
    for (int g = 0; g < PIX / 256; ++g)
        atomicAdd(&lc[idxreg[g]], 1.0f);

    for (int half = 0; half < 2; ++half) {
        for (int i = tid; i < 128 * Kn; i += 256) ls[i] = 0.0f;
        __syncthreads();
#pragma unroll
        for (int g = 0; g < PIX / 256; ++g) {
            const int p = hw0 + g * 256 + tid;
            const int idx = idxreg[g];
            for (int cc = 0; cc < 128; ++cc) {
                const int c = half * 128 + cc;
                float v = x[((size_t)b * Cn + c) * HWn + p];
                atomicAdd(&ls[cc * Kn + idx], v);   // ds_add_f32, bank = idx
            }
        }
        __syncthreads();
        for (int i = tid; i < 128 * Kn; i += 256) {
            const int cc = i >> 6, kk = i & 63;
            float v = ls[i];
            if (v != 0.0f)
                atomicAdd(&sums[((size_t)b * Kn + kk) * Cn + half * 128 + cc], v);
        }
        __syncthreads();
    }
    if (tid < Kn) {
        float v = lc[tid];
        if (v != 0.0f) atomicAdd(&counts[b * Kn + tid], v);
    }
}

// ---------------------------------------------------------------- means
__global__ __launch_bounds__(256) void means_kernel(const float* __restrict__ sums,
                                                    const float* __restrict__ counts,
                                                    float* __restrict__ means) {
    const int r = blockIdx.x;                    // b*K + k
    float cnt = counts[r];
    float denom = cnt + (cnt == 0.0f ? 1.0f : 0.0f);
    means[(size_t)r * Cn + threadIdx.x] = sums[(size_t)r * Cn + threadIdx.x] / denom;
}

// ---------------------------------------------------------------- M = W W^T
__global__ __launch_bounds__(256) void gram_kernel(const float* __restrict__ Wm,
                                                   float* __restrict__ Mmat) {
    __shared__ float wr[Cn];
    const int i = blockIdx.x, j = threadIdx.x;
    wr[j] = Wm[(size_t)i * Cn + j];
    __syncthreads();
    float acc = 0.0f;
    for (int c = 0; c < Cn; ++c) acc += wr[c] * Wm[(size_t)j * Cn + c];
    Mmat[(size_t)i * Cn + j] = acc;
}

// -------------------------------------- P = means * M ; qdiag = rowdot(P,means)
__global__ __launch_bounds__(256) void pq_kernel(const float* __restrict__ means,
                                                 const float* __restrict__ Mmat,
                                                 float* __restrict__ P,
                                                 float* __restrict__ qdiag) {
    __shared__ float ml[Cn];
    __shared__ float red[256];
    const int r = blockIdx.x, tid = threadIdx.x;
    ml[tid] = means[(size_t)r * Cn + tid];
    __syncthreads();
    float acc = 0.0f;
    for (int d = 0; d < Cn; ++d) acc += ml[d] * Mmat[(size_t)d * Cn + tid];
    P[(size_t)r * Cn + tid] = acc;
    red[tid] = acc * ml[tid];
    __syncthreads();
    for (int s = 128; s > 0; s >>= 1) {
        if (tid < s) red[tid] += red[tid + s];
        __syncthreads();
    }
    if (tid == 0) qdiag[r] = red[0];
}

// ---------------------------------------------- adj row (implicit) + agg row
__global__ __launch_bounds__(256) void adjagg_kernel(const float* __restrict__ means,
                                                     const float* __restrict__ P,
                                                     const float* __restrict__ qdiag,
                                                     float* __restrict__ agg) {
    __shared__ float pm[Cn];
    __shared__ float adjs[Kn];
    const int tid = threadIdx.x;
    const int b = blockIdx.x >> 6;
    const int i = blockIdx.x & 63;
    pm[tid] = P[(size_t)blockIdx.x * Cn + tid];
    __syncthreads();
    if (tid < Kn) {
        const int j = tid;
        const float* mj = means + ((size_t)b * Kn + j) * Cn;
        float g = 0.0f;
        for (int d = 0; d < Cn; ++d) g += pm[d] * mj[d];
        float quad = qdiag[b * Kn + i] + qdiag[b * Kn + j] - 2.0f * g;
        adjs[j] = (j == i) ? 0.0f : expf(-quad);
    }
    __syncthreads();
    float acc = 0.0f;
    for (int j = 0; j < Kn; ++j)
        acc += adjs[j] * means[((size_t)b * Kn + j) * Cn + tid];
    agg[(size_t)blockIdx.x * Cn + tid] = acc;
}

// ------------------------------------------- cagg[b,k,o] = conv_w[o,:]·agg[b,k,:]
__global__ __launch_bounds__(256) void cagg_kernel(const float* __restrict__ agg,
                                                   const float* __restrict__ cw,
                                                   float* __restrict__ cagg) {
    __shared__ float al[Cn];
    const int tid = threadIdx.x;
    al[tid] = agg[(size_t)blockIdx.x * Cn + tid];
    __syncthreads();
    float acc = 0.0f;
    for (int c = 0; c < Cn; ++c) acc += al[c] * cw[(size_t)tid * Cn + c];
    cagg[(size_t)blockIdx.x * Cn + tid] = acc;
}

// ---------------------------------------------------------------- TDM helpers
#define KCH 16
#define NCHUNK (Cn / KCH)     // 16
#define APITCH 17             // 16 dw row + 1 dw pad (17 coprime 64 -> no conflicts)
#define BPITCH 136            // 128 dw row + 8 dw pad (upper half-wave shifted 16 banks)

#if HAVE_TDM
// D# group 0: count=1, lds_addr, 57-bit global addr, type=2 ("image")
static __device__ __forceinline__ v4u tdm_group0(unsigned int lds_byte_off,
                                                 const void* gptr) {
    unsigned long long ga = (unsigned long long)(uintptr_t)gptr;
    v4u g;
    g.x = 1u;
    g.y = lds_byte_off;
    g.z = (unsigned int)ga;
    g.w = ((unsigned int)(ga >> 32) & 0x01FFFFFFu) | 0x80000000u;
    return g;
}
// D# group 1: data_size=4B, pad_enable, 2-D tile (tile_dim2=0)
static __device__ __forceinline__ v8i_ tdm_group1(unsigned int td0, unsigned int td1,
                                                  unsigned int tl0, unsigned int tl1,
                                                  unsigned int stride0,
                                                  unsigned int pad_int,
                                                  unsigned int pad_amt) {
    v8i_ g;
    g[0] = (int)((2u << 16) | (1u << 20) | (pad_int << 22) | (pad_amt << 25));
    g[1] = (int)((td0 & 0xFFFFu) << 16);
    g[2] = (int)(((td0 >> 16) & 0xFFFFu) | ((td1 & 0xFFFFu) << 16));
    g[3] = (int)(((td1 >> 16) & 0xFFFFu) | (tl0 << 16));
    g[4] = (int)(tl1 & 0xFFFFu);
    g[5] = (int)stride0;          // tensor_dim0_stride (fits 32b here)
    g[6] = 0;
    g[7] = 0;
    return g;
}
#endif

// ---------------- main 1x1-conv GEMM: TDM double-buffered LDS + WMMA f32 ------
// out[b,o,n] = sum_c cw[o,c] * x[b,c,n]  +  cagg[b, index[b,n], o]
// Block: 8 waves, 4(M) x 2(N); wave tile 32(o) x 64(n); block tile 128 x 128.
__global__ __launch_bounds__(256) void conv_wmma_kernel(const float* __restrict__ x,
                                                        const int* __restrict__ index,
                                                        const float* __restrict__ cw,
                                                        const float* __restrict__ cagg,
                                                        float* __restrict__ out) {
    __shared__ float Ash[2][128 * APITCH];  // 2 x 8704 B
    __shared__ float Bsh[2][KCH * BPITCH];  // 2 x 8704 B

    const int b = blockIdx.z;
    const int oBlk = blockIdx.y * 128;
    const int nBlk = blockIdx.x * 128;
    const int tid = threadIdx.x;
    const int lane = tid & 31;
    const int wid = tid >> 5;
    const int wm = wid >> 1;            // 0..3
    const int wn = wid & 1;             // 0..1
    const int mBase = wm * 32;
    const int nBase = wn * 64;
    const int lhalf = lane >> 4;        // selects K pair
    const int l15 = lane & 15;

    const float* xb = x + (size_t)b * Cn * HWn;

    v8f acc[2][4] = {};

#if HAVE_TDM
    const bool wlead = __builtin_amdgcn_readfirstlane(tid >> 5) == 0;  // scalar branch
    v4i_ z4 = {};
    v8i_ z8 = {};
    if (wlead) {
        __builtin_amdgcn_tensor_load_to_lds(
            tdm_group0((unsigned int)(uintptr_t)&Ash[0][0], cw + (size_t)oBlk * Cn),
            tdm_group1(Cn, Cn, KCH, 128, Cn, 3u, 0u), z4, z4, z8, 0);
        __builtin_amdgcn_tensor_load_to_lds(
            tdm_group0((unsigned int)(uintptr_t)&Bsh[0][0], xb + nBlk),
            tdm_group1(HWn, Cn, 128, KCH, HWn, 6u, 7u), z4, z4, z8, 0);
    }
#endif

    for (int ch = 0; ch < NCHUNK; ++ch) {
        const int cur = ch & 1;
#if HAVE_TDM
        if (wlead) {
            if (ch + 1 < NCHUNK) {
                const int kn = (ch + 1) * KCH;
                __builtin_amdgcn_tensor_load_to_lds(
                    tdm_group0((unsigned int)(uintptr_t)&Ash[cur ^ 1][0],
                               cw + (size_t)oBlk * Cn + kn),
                    tdm_group1(Cn, Cn, KCH, 128, Cn, 3u, 0u), z4, z4, z8, 0);
                __builtin_amdgcn_tensor_load_to_lds(
                    tdm_group0((unsigned int)(uintptr_t)&Bsh[cur ^ 1][0],
                               xb + (size_t)kn * HWn + nBlk),
                    tdm_group1(HWn, Cn, 128, KCH, HWn, 6u, 7u), z4, z4, z8, 0);
                __builtin_amdgcn_s_wait_tensorcnt(2);  // current chunk's 2 DMAs done
            } else {
                __builtin_amdgcn_s_wait_tensorcnt(0);
            }
        }
        __syncthreads();
#else
        // Fallback: cooperative synchronous staging
        for (int i = tid; i < 128 * KCH; i += 256) {
            const int m = i / KCH, kk = i % KCH;
            Ash[cur][m * APITCH + kk] = cw[(size_t)(oBlk + m) * Cn + ch * KCH + kk];
        }
        for (int i = tid; i < KCH * 128; i += 256) {
            const int r = i >> 7, cc = i & 127;
            Bsh[cur][r * BPITCH + cc] = xb[(size_t)(ch * KCH + r) * HWn + nBlk + cc];
        }
        __syncthreads();
#endif
        const float* Ab = &Ash[cur][0];
        const float* Bb = &Bsh[cur][0];
#pragma unroll
        for (int ks = 0; ks < KCH; ks += 4) {
            const int kk = ks + lhalf * 2;
            // A frags (16x4 f32): lane<16 -> K=kk,kk+1 ; lane>=16 -> K=kk+2,kk+3
            v2f a[2];
            a[0][0] = Ab[(mBase + l15) * APITCH + kk];
            a[0][1] = Ab[(mBase + l15) * APITCH + kk + 1];
            a[1][0] = Ab[(mBase + 16 + l15) * APITCH + kk];
            a[1][1] = Ab[(mBase + 16 + l15) * APITCH + kk + 1];
            // B frags (4x16 f32): mirrored; column N = l15
            v2f bf[4];
#pragma unroll
            for (int nt = 0; nt < 4; ++nt) {
                bf[nt][0] = Bb[kk * BPITCH + nBase + nt * 16 + l15];
                bf[nt][1] = Bb[(kk + 1) * BPITCH + nBase + nt * 16 + l15];
            }
#pragma unroll
            for (int mt = 0; mt < 2; ++mt)
#pragma unroll
                for (int nt = 0; nt < 4; ++nt)
                    acc[mt][nt] = __builtin_amdgcn_wmma_f32_16x16x4_f32(
                        false, a[mt], false, bf[nt], (short)0, acc[mt][nt], false, false);
        }
        __syncthreads();
    }

    // Epilogue: C/D layout — lane<16: M=v, N=lane ; lane>=16: M=8+v, N=lane-16
    const int oW = oBlk + mBase;
    const int nW = nBlk + nBase;
#pragma unroll
    for (int nt = 0; nt < 4; ++nt) {
        const int n = nW + nt * 16 + l15;
        const int idx = index[b * HWn + n];
        const float* cg = cagg + ((size_t)b * Kn + idx) * Cn;
#pragma unroll
        for (int mt = 0; mt < 2; ++mt) {
#pragma unroll
            for (int v = 0; v < 8; ++v) {
                const int m = oW + mt * 16 + lhalf * 8 + v;
                out[((size_t)b * Cn + m) * HWn + n] = acc[mt][nt][v] + cg[m];
            }
        }
    }
}

// ---------------------------------------------------------------- launch
extern "C" void kernel_launch(void* const* d_in, const int* in_sizes, int n_in,
                              void* d_out, int out_size, void* d_ws, size_t ws_size,
                              hipStream_t stream) {
    const float* x      = (const float*)d_in[0];
    const int*   index  = (const int*)d_in[1];
    const float* Wm     = (const float*)d_in[2];
    const float* conv_w = (const float*)d_in[3];
    float* out = (float*)d_out;

    float* ws     = (float*)d_ws;
    float* sums   = ws;                    // B*K*C
    float* counts = sums   + Bn * Kn * Cn; // B*K
    float* means  = counts + Bn * Kn;      // B*K*C
    float* Mmat   = means  + Bn * Kn * Cn; // C*C
    float* P      = Mmat   + Cn * Cn;      // B*K*C
    float* qdiag  = P      + Bn * Kn * Cn; // B*K
    float* agg    = qdiag  + Bn * Kn;      // B*K*C
    float* cagg   = agg    + Bn * Kn * Cn; // B*K*C

    const int zn = Bn * Kn * Cn + Bn * Kn;
    zero_kernel<<<(zn + 255) / 256, 256, 0, stream>>>(sums, zn);

    scatter_kernel<<<dim3(Bn, HWn / PIX), 256, 0, stream>>>(x, index, sums, counts);
    means_kernel<<<Bn * Kn, 256, 0, stream>>>(sums, counts, means);
    gram_kernel<<<Cn, 256, 0, stream>>>(Wm, Mmat);
    pq_kernel<<<Bn * Kn, 256, 0, stream>>>(means, Mmat, P, qdiag);
    adjagg_kernel<<<Bn * Kn, 256, 0, stream>>>(means, P, qdiag, agg);
    cagg_kernel<<<Bn * Kn, 256, 0, stream>>>(agg, conv_w, cagg);

    conv_wmma_kernel<<<dim3(HWn / 128, Cn / 128, Bn), 256, 0, stream>>>(
        x, index, conv_w, cagg, out);
}